// sGAT_40948218200367
// MI455X (gfx1250) — compile-verified
//
#include <hip/hip_runtime.h>
#include <hip/hip_bf16.h>

#define NN 50000
#define NE 640000
#define NG 256
#define HID 128

typedef float v2f __attribute__((ext_vector_type(2)));
typedef float v8f __attribute__((ext_vector_type(8)));

__device__ __forceinline__ unsigned ordf(float f) {
    unsigned u = __float_as_uint(f);
    return (u >> 31) ? ~u : (u | 0x80000000u);
}
__device__ __forceinline__ float unordf(unsigned u) {
    return (u >> 31) ? __uint_as_float(u & 0x7FFFFFFFu) : __uint_as_float(~u);
}

// --- per-layer state init: acc = 0, denom = 0, amax = ord(-inf) ------------
__global__ void init_kernel(float* __restrict__ acc, float* __restrict__ denom,
                            unsigned* __restrict__ amax) {
    int i = blockIdx.x * blockDim.x + threadIdx.x;
    if (i < NN * HID) acc[i] = 0.f;
    if (i < NN * 2) { denom[i] = 0.f; amax[i] = 0x007FFFFFu; /* ordf(-inf) */ }
}

// --- node GEMM: xh[N,256] = hin[N,128] @ W[128,256] + bias, via f32 WMMA ---
// one wave per 16x16 output tile; 32 chained v_wmma_f32_16x16x4_f32 (K=128)
__global__ void gemm_xh_kernel(const float* __restrict__ A, const float* __restrict__ W,
                               const float* __restrict__ bias_l, float* __restrict__ xh) {
    int wave = (blockIdx.x * blockDim.x + threadIdx.x) >> 5;   // tile id, exact 50000
    int lane = threadIdx.x & 31;
    int mt = wave >> 4, nt = wave & 15;
    int mn = lane & 15, half = lane >> 4;                      // mn = M row (A) / N col (B,D)
    const float* Arow = A + (size_t)(mt * 16 + mn) * 128 + 2 * half;
    const float* Bcol = W + (size_t)(2 * half) * 256 + nt * 16 + mn;
    v8f c = {};
#pragma unroll 4
    for (int k = 0; k < 128; k += 4) {
        v2f a; a.x = Arow[k];                 a.y = Arow[k + 1];
        v2f b; b.x = Bcol[(size_t)k * 256];   b.y = Bcol[(size_t)k * 256 + 256];
        c = __builtin_amdgcn_wmma_f32_16x16x4_f32(false, a, false, b, (short)0, c,
                                                  false, false);
    }
    float bn = bias_l[nt * 16 + mn];
    float* out = xh + (size_t)(mt * 16 + 8 * half) * 256 + nt * 16 + mn;
#pragma unroll
    for (int r = 0; r < 8; ++r) out[(size_t)r * 256] = c[r] + bn;
}

// --- fold attention vector onto nodes: aL/aR[n,h] --------------------------
__global__ void att_proj_kernel(const float* __restrict__ xh, const float* __restrict__ att_l,
                                float* __restrict__ aL, float* __restrict__ aR) {
    int n = (blockIdx.x * blockDim.x + threadIdx.x) >> 5;
    int lane = threadIdx.x & 31;
    if (n >= NN) return;
    const float4* xr = (const float4*)(xh + (size_t)n * 256);
#pragma unroll
    for (int h = 0; h < 2; ++h) {
        float4 xv = xr[h * 32 + lane];
        const float4* a0 = (const float4*)(att_l + h * 260);
        float4 av = a0[lane];          // att[h][0:128]
        float4 bv = a0[32 + lane];     // att[h][128:256]
        float pl = xv.x * av.x + xv.y * av.y + xv.z * av.z + xv.w * av.w;
        float pr = xv.x * bv.x + xv.y * bv.y + xv.z * bv.z + xv.w * bv.w;
        for (int off = 16; off; off >>= 1) {
            pl += __shfl_xor(pl, off);
            pr += __shfl_xor(pr, off);
        }
        if (lane == 0) { aL[n * 2 + h] = pl; aR[n * 2 + h] = pr; }
    }
}

// --- per-edge: eh, alpha (leaky), segment-max; also next-layer e_out -------
__global__ void edge_alpha_kernel(const int* __restrict__ ei, const float* __restrict__ ein,
                                  const float* __restrict__ linE_l, const float* __restrict__ att_l,
                                  const float* __restrict__ aL, const float* __restrict__ aR,
                                  float* __restrict__ alphaBuf, unsigned* __restrict__ amax,
                                  float* __restrict__ ebuf) {
    int e = blockIdx.x * blockDim.x + threadIdx.x;
    if (e >= NE) return;
    int s = ei[e], d = ei[NE + e];
    float4 ea = ((const float4*)ein)[e];
    float eh[2][4];
#pragma unroll
    for (int h = 0; h < 2; ++h)
#pragma unroll
        for (int j = 0; j < 4; ++j)
            eh[h][j] = ea.x * linE_l[h * 4 + j]      + ea.y * linE_l[8 + h * 4 + j]
                     + ea.z * linE_l[16 + h * 4 + j] + ea.w * linE_l[24 + h * 4 + j];
#pragma unroll
    for (int h = 0; h < 2; ++h) {
        float al = aL[s * 2 + h] + aR[d * 2 + h];
#pragma unroll
        for (int j = 0; j < 4; ++j) al += att_l[h * 260 + 256 + j] * eh[h][j];
        al = al >= 0.f ? al : 0.2f * al;           // leaky relu, slope 0.2
        alphaBuf[e * 2 + h] = al;
        atomicMax(&amax[d * 2 + h], ordf(al));
    }
    float4 eo;
    eo.x = fmaxf(0.f, 0.5f * (eh[0][0] + eh[1][0]));
    eo.y = fmaxf(0.f, 0.5f * (eh[0][1] + eh[1][1]));
    eo.z = fmaxf(0.f, 0.5f * (eh[0][2] + eh[1][2]));
    eo.w = fmaxf(0.f, 0.5f * (eh[0][3] + eh[1][3]));
    ((float4*)ebuf)[e] = eo;
}

// --- per-edge: ex = exp(alpha - max), denom += ex (in-place over alphaBuf) -
__global__ void edge_exp_kernel(const int* __restrict__ ei, float* __restrict__ alphaBuf,
                                const unsigned* __restrict__ amax, float* __restrict__ denom) {
    int e = blockIdx.x * blockDim.x + threadIdx.x;
    if (e >= NE) return;
    int d = ei[NE + e];
#pragma unroll
    for (int h = 0; h < 2; ++h) {
        float a = alphaBuf[e * 2 + h];
        float ex = __expf(a - unordf(amax[d * 2 + h]));
        alphaBuf[e * 2 + h] = ex;
        atomicAdd(&denom[d * 2 + h], ex);
    }
}

// --- per-edge message scatter, heads pre-summed (mean taken later) ---------
__global__ void edge_msg_kernel(const int* __restrict__ ei, const float* __restrict__ exBuf,
                                const float* __restrict__ denom, const float* __restrict__ xh,
                                float* __restrict__ acc) {
    int e = (blockIdx.x * blockDim.x + threadIdx.x) >> 5;
    int lane = threadIdx.x & 31;
    if (e >= NE) return;
    int s = ei[e], d = ei[NE + e];
    float w0 = exBuf[e * 2]     / (denom[d * 2]     + 1e-16f);
    float w1 = exBuf[e * 2 + 1] / (denom[d * 2 + 1] + 1e-16f);
    const float4* xr = (const float4*)(xh + (size_t)s * 256);
    float4 x0 = xr[lane], x1 = xr[32 + lane];
    float4 v;
    v.x = w0 * x0.x + w1 * x1.x;
    v.y = w0 * x0.y + w1 * x1.y;
    v.z = w0 * x0.z + w1 * x1.z;
    v.w = w0 * x0.w + w1 * x1.w;
    float* ap = acc + (size_t)d * 128 + lane * 4;
    atomicAdd(ap + 0, v.x);
    atomicAdd(ap + 1, v.y);
    atomicAdd(ap + 2, v.z);
    atomicAdd(ap + 3, v.w);
}

// --- h_next = relu(0.5*(acc + xh_head0 + xh_head1)) ------------------------
__global__ void node_final_kernel(const float* __restrict__ acc, const float* __restrict__ xh,
                                  float* __restrict__ hbuf) {
    int idx = blockIdx.x * blockDim.x + threadIdx.x;  // over NN*32 float4s
    if (idx >= NN * 32) return;
    int n = idx >> 5, q = idx & 31;
    float4 a  = ((const float4*)acc)[idx];
    float4 x0 = ((const float4*)xh)[n * 64 + q];
    float4 x1 = ((const float4*)xh)[n * 64 + 32 + q];
    float4 r;
    r.x = fmaxf(0.f, 0.5f * (a.x + x0.x + x1.x));
    r.y = fmaxf(0.f, 0.5f * (a.y + x0.y + x1.y));
    r.z = fmaxf(0.f, 0.5f * (a.z + x0.z + x1.z));
    r.w = fmaxf(0.f, 0.5f * (a.w + x0.w + x1.w));
    ((float4*)hbuf)[idx] = r;
}

__global__ void out_init_kernel(float* __restrict__ y, const float* __restrict__ bf) {
    int g = threadIdx.x;
    if (g < NG) y[g] = bf[0];
}

// --- y[batch[n]] += dot(h[n], Wf) ------------------------------------------
__global__ void pool_kernel(const float* __restrict__ hbuf, const int* __restrict__ batch,
                            const float* __restrict__ Wf, float* __restrict__ y) {
    int n = (blockIdx.x * blockDim.x + threadIdx.x) >> 5;
    int lane = threadIdx.x & 31;
    if (n >= NN) return;
    float4 hv = ((const float4*)(hbuf + (size_t)n * 128))[lane];
    float4 wv = ((const float4*)Wf)[lane];
    float p = hv.x * wv.x + hv.y * wv.y + hv.z * wv.z + hv.w * wv.w;
    for (int off = 16; off; off >>= 1) p += __shfl_xor(p, off);
    if (lane == 0) atomicAdd(&y[batch[n]], p);
}

extern "C" void kernel_launch(void* const* d_in, const int* in_sizes, int n_in,
                              void* d_out, int out_size, void* d_ws, size_t ws_size,
                              hipStream_t stream) {
    const float* x         = (const float*)d_in[0];
    const int*   ei        = (const int*)d_in[1];
    const float* edge_attr = (const float*)d_in[2];
    const int*   batch     = (const int*)d_in[3];
    const float* linN      = (const float*)d_in[4];
    const float* linE      = (const float*)d_in[5];
    const float* att       = (const float*)d_in[6];
    const float* bias      = (const float*)d_in[7];
    const float* Wf        = (const float*)d_in[8];
    const float* bf        = (const float*)d_in[9];
    float* y = (float*)d_out;

    float* ws       = (float*)d_ws;
    float* xh       = ws;                     // NN*256 = 12.8M
    float* acc      = xh + 12800000;          // NN*128 = 6.4M
    float* hbuf     = acc + 6400000;          // NN*128 = 6.4M
    float* ebuf     = hbuf + 6400000;         // NE*4   = 2.56M
    float* alphaBuf = ebuf + 2560000;         // NE*2   = 1.28M
    float* aL       = alphaBuf + 1280000;     // NN*2
    float* aR       = aL + 100000;            // NN*2
    unsigned* amax  = (unsigned*)(aR + 100000); // NN*2
    float* denom    = (float*)(amax + 100000);  // NN*2   -> ~119 MB total

    for (int l = 0; l < 3; ++l) {
        const float* hin    = (l == 0) ? x : hbuf;
        const float* ein    = (l == 0) ? edge_attr : ebuf;
        const float* linN_l = linN + (size_t)l * 128 * 256;
        const float* linE_l = linE + l * 32;
        const float* att_l  = att + l * 520;
        const float* bias_l = bias + l * 256;

        init_kernel<<<25000, 256, 0, stream>>>(acc, denom, amax);
        gemm_xh_kernel<<<12500, 128, 0, stream>>>(hin, linN_l, bias_l, xh);
        att_proj_kernel<<<12500, 128, 0, stream>>>(xh, att_l, aL, aR);
        edge_alpha_kernel<<<2500, 256, 0, stream>>>(ei, ein, linE_l, att_l, aL, aR,
                                                    alphaBuf, amax, ebuf);
        edge_exp_kernel<<<2500, 256, 0, stream>>>(ei, alphaBuf, amax, denom);
        edge_msg_kernel<<<160000, 128, 0, stream>>>(ei, alphaBuf, denom, xh, acc);
        node_final_kernel<<<6250, 256, 0, stream>>>(acc, xh, hbuf);
    }
    out_init_kernel<<<1, 256, 0, stream>>>(y, bf);
    pool_kernel<<<12500, 128, 0, stream>>>(hbuf, batch, Wf, y);
}